// S6_21285857919289
// MI455X (gfx1250) — compile-verified
//
#include <hip/hip_runtime.h>
#include <hip/hip_bf16.h>

typedef float v2f __attribute__((ext_vector_type(2)));
typedef float v8f __attribute__((ext_vector_type(8)));

// Problem constants (from reference setup_inputs)
#define B_  2
#define L_  2048
#define D_  512
#define N_  64
#define R_  32
#define E_  160          // r + 2n
#define BL_ (B_ * L_)    // 4096
#define G_  32           // chunks along l
#define T_  (L_ / G_)    // 64 steps per chunk

// ---------------------------------------------------------------------------
// GEMM 1: x_dbl[bl, e] = sum_k x[bl, k] * W_xproj[e, k]   (f32 WMMA 16x16x4)
// One wave per 16x16 output tile. 256 row-tiles x 10 e-tiles = 2560 waves.
// ---------------------------------------------------------------------------
__global__ void __launch_bounds__(256)
s6_xproj_gemm(const float* __restrict__ x, const float* __restrict__ W,
              float* __restrict__ xdbl) {
  const int wave  = blockIdx.x * (blockDim.x >> 5) + (threadIdx.x >> 5);
  const int lane  = threadIdx.x & 31;
  const int rt    = wave / 10;          // row tile (bl)
  const int et    = wave % 10;          // e tile
  const int laneM = lane & 15;
  const int laneH = lane >> 4;          // 0: K,K+1 ; 1: K+2,K+3

  const float* arow = x + (size_t)(rt * 16 + laneM) * D_ + 2 * laneH;
  const float* brow = W + (size_t)(et * 16 + laneM) * D_ + 2 * laneH;

  v8f c = {};
  for (int k = 0; k < D_; k += 4) {
    v2f a = *(const v2f*)(arow + k);
    v2f b = *(const v2f*)(brow + k);
    c = __builtin_amdgcn_wmma_f32_16x16x4_f32(false, a, false, b,
                                              (short)0, c, false, false);
  }
  // C/D layout: VGPR j -> M = j + 8*laneH, N = laneM
  const int col = et * 16 + laneM;
#pragma unroll
  for (int j = 0; j < 8; ++j) {
    const int row = rt * 16 + j + 8 * laneH;
    xdbl[(size_t)row * E_ + col] = c[j];
  }
}

// ---------------------------------------------------------------------------
// GEMM 2: delta[bl, d] = softplus( sum_r x_dbl[bl, r] * W_dt[d, r] + b_dt[d] )
// K = 32 -> 8 WMMA steps. 256 row-tiles x 32 d-tiles = 8192 waves.
// ---------------------------------------------------------------------------
__global__ void __launch_bounds__(256)
s6_delta_gemm(const float* __restrict__ xdbl, const float* __restrict__ Wdt,
              const float* __restrict__ bdt, float* __restrict__ delta) {
  const int wave  = blockIdx.x * (blockDim.x >> 5) + (threadIdx.x >> 5);
  const int lane  = threadIdx.x & 31;
  const int rt    = wave >> 5;          // row tile (bl), 256 of them
  const int nt    = wave & 31;          // d tile, 32 of them
  const int laneM = lane & 15;
  const int laneH = lane >> 4;

  const float* arow = xdbl + (size_t)(rt * 16 + laneM) * E_ + 2 * laneH;
  const float* brow = Wdt  + (size_t)(nt * 16 + laneM) * R_ + 2 * laneH;

  v8f c = {};
#pragma unroll
  for (int k = 0; k < R_; k += 4) {
    v2f a = *(const v2f*)(arow + k);
    v2f b = *(const v2f*)(brow + k);
    c = __builtin_amdgcn_wmma_f32_16x16x4_f32(false, a, false, b,
                                              (short)0, c, false, false);
  }
  const int col  = nt * 16 + laneM;     // d index
  const float bi = bdt[col];
#pragma unroll
  for (int j = 0; j < 8; ++j) {
    const int row = rt * 16 + j + 8 * laneH;
    float z = c[j] + bi;
    float sp = (z > 20.0f) ? z : __logf(1.0f + __expf(z));
    delta[(size_t)row * D_ + col] = sp;
  }
}

// ---------------------------------------------------------------------------
// Pass 1: per (b,d,chunk) wave compute diagonal transition product P[n] and
// particular end-state Hhat[n] (h0 = 0). Lane owns n = {2*lane, 2*lane+1}.
// ---------------------------------------------------------------------------
__global__ void __launch_bounds__(256)
s6_scan_chunks(const float* __restrict__ xdbl, const float* __restrict__ delta,
               const float* __restrict__ x, const float* __restrict__ Alog,
               float* __restrict__ Pout, float* __restrict__ Hout) {
  const int wave = blockIdx.x * (blockDim.x >> 5) + (threadIdx.x >> 5);
  const int lane = threadIdx.x & 31;
  const int g  = wave % G_;
  const int bd = wave / G_;
  const int dd = bd % D_;
  const int bb = bd / D_;
  const int n0 = 2 * lane;

  v2f al = *(const v2f*)(Alog + (size_t)dd * N_ + n0);
  v2f A;  A.x = -__expf(al.x);  A.y = -__expf(al.y);     // A = -exp(A_log)

  v2f h = {0.0f, 0.0f};
  v2f P = {1.0f, 1.0f};
  const int t0 = g * T_;
  for (int tt = 0; tt < T_; ++tt) {
    const size_t row = (size_t)bb * L_ + (t0 + tt);
    const float dlt = delta[row * D_ + dd];
    const float xv  = x[row * D_ + dd];
    v2f Bv = *(const v2f*)(xdbl + row * E_ + R_ + n0);
    if (tt + 8 < T_)
      __builtin_prefetch(xdbl + (row + 8) * E_ + R_ + n0, 0, 1);
    v2f a;  a.x = __expf(dlt * A.x);  a.y = __expf(dlt * A.y);
    const float du = dlt * xv;
    h.x = a.x * h.x + du * Bv.x;
    h.y = a.y * h.y + du * Bv.y;
    P.x *= a.x;  P.y *= a.y;
  }
  const size_t base = ((size_t)bd * G_ + g) * N_ + n0;
  *(v2f*)(Pout + base) = P;
  *(v2f*)(Hout + base) = h;
}

// ---------------------------------------------------------------------------
// Combine: per (b,d) wave, sequentially chain G chunks: h0[g] known, then
// h0[g+1] = P[g] * h0[g] + Hhat[g].
// ---------------------------------------------------------------------------
__global__ void __launch_bounds__(256)
s6_combine(const float* __restrict__ Pin, const float* __restrict__ Hin,
           float* __restrict__ h0out) {
  const int bd   = blockIdx.x * (blockDim.x >> 5) + (threadIdx.x >> 5);
  const int lane = threadIdx.x & 31;
  const int n0   = 2 * lane;

  v2f h = {0.0f, 0.0f};
  for (int g = 0; g < G_; ++g) {
    const size_t base = ((size_t)bd * G_ + g) * N_ + n0;
    *(v2f*)(h0out + base) = h;
    v2f P = *(const v2f*)(Pin + base);
    v2f H = *(const v2f*)(Hin + base);
    h.x = P.x * h.x + H.x;
    h.y = P.y * h.y + H.y;
  }
}

// ---------------------------------------------------------------------------
// Pass 2: per (b,d,chunk) wave, replay chunk from true h0, emit
// y[b,t,d] = sum_n C[b,t,n] * h[n]  + x*D   (wave shuffle reduction over n).
// ---------------------------------------------------------------------------
__global__ void __launch_bounds__(256)
s6_scan_emit(const float* __restrict__ xdbl, const float* __restrict__ delta,
             const float* __restrict__ x, const float* __restrict__ Alog,
             const float* __restrict__ Dp, const float* __restrict__ h0in,
             float* __restrict__ y) {
  const int wave = blockIdx.x * (blockDim.x >> 5) + (threadIdx.x >> 5);
  const int lane = threadIdx.x & 31;
  const int g  = wave % G_;
  const int bd = wave / G_;
  const int dd = bd % D_;
  const int bb = bd / D_;
  const int n0 = 2 * lane;

  v2f al = *(const v2f*)(Alog + (size_t)dd * N_ + n0);
  v2f A;  A.x = -__expf(al.x);  A.y = -__expf(al.y);
  const float Dv = Dp[dd];

  v2f h = *(const v2f*)(h0in + ((size_t)bd * G_ + g) * N_ + n0);

  const int t0 = g * T_;
  for (int tt = 0; tt < T_; ++tt) {
    const size_t row = (size_t)bb * L_ + (t0 + tt);
    const float dlt = delta[row * D_ + dd];
    const float xv  = x[row * D_ + dd];
    v2f Bv = *(const v2f*)(xdbl + row * E_ + R_ + n0);        // B
    v2f Cv = *(const v2f*)(xdbl + row * E_ + R_ + N_ + n0);   // C
    if (tt + 8 < T_)
      __builtin_prefetch(xdbl + (row + 8) * E_ + R_ + n0, 0, 1);
    v2f a;  a.x = __expf(dlt * A.x);  a.y = __expf(dlt * A.y);
    const float du = dlt * xv;
    h.x = a.x * h.x + du * Bv.x;
    h.y = a.y * h.y + du * Bv.y;
    float part = h.x * Cv.x + h.y * Cv.y;
#pragma unroll
    for (int off = 16; off >= 1; off >>= 1)
      part += __shfl_xor(part, off, 32);
    if (lane == 0)
      y[row * D_ + dd] = part + xv * Dv;
  }
}

// ---------------------------------------------------------------------------
extern "C" void kernel_launch(void* const* d_in, const int* in_sizes, int n_in,
                              void* d_out, int out_size, void* d_ws, size_t ws_size,
                              hipStream_t stream) {
  const float* x      = (const float*)d_in[0];  // (2,2048,512)
  const float* Wxproj = (const float*)d_in[1];  // (160,512)
  const float* Wdt    = (const float*)d_in[2];  // (512,32)
  const float* bdt    = (const float*)d_in[3];  // (512,)
  const float* Alog   = (const float*)d_in[4];  // (512,64)
  const float* Dp     = (const float*)d_in[5];  // (512,)
  float*       y      = (float*)d_out;          // (2,2048,512)

  float* ws    = (float*)d_ws;
  float* xdbl  = ws;                                   // 4096*160
  float* delta = xdbl  + (size_t)BL_ * E_;             // 4096*512
  float* Pws   = delta + (size_t)BL_ * D_;             // 1024*32*64
  float* Hws   = Pws   + (size_t)B_ * D_ * G_ * N_;
  float* h0ws  = Hws   + (size_t)B_ * D_ * G_ * N_;

  // 2560 tile-waves, 8 waves / 256-thread block
  s6_xproj_gemm<<<320, 256, 0, stream>>>(x, Wxproj, xdbl);
  // 8192 tile-waves
  s6_delta_gemm<<<1024, 256, 0, stream>>>(xdbl, Wdt, bdt, delta);
  // 32768 chunk-waves
  s6_scan_chunks<<<4096, 256, 0, stream>>>(xdbl, delta, x, Alog, Pws, Hws);
  // 1024 channel-waves
  s6_combine<<<128, 256, 0, stream>>>(Pws, Hws, h0ws);
  // 32768 chunk-waves
  s6_scan_emit<<<4096, 256, 0, stream>>>(xdbl, delta, x, Alog, Dp, h0ws, y);
}